// CurvatureComputer_66529043415118
// MI455X (gfx1250) — compile-verified
//
#include <hip/hip_runtime.h>
#include <hip/hip_bf16.h>
#include <stdint.h>

// Problem geometry (fixed by setup_inputs): depth (32,1,512,512) f32 -> out (32,10,512,512) f32
#define IMG_W 512
#define IMG_H 512
#define NBATCH 32
#define TX 64              // output tile width
#define TY 16              // output tile height
#define DROWS 20           // depth tile rows   (TY + 4)
#define DST   68           // depth tile stride (TX + 4), floats, packed (TDM writes rows packed)
#define GROWS 18           // grad tile rows    (TY + 2)
#define GST   66           // grad tile stride  (TX + 2)
#define NTHREADS 256

typedef uint32_t v4u __attribute__((ext_vector_type(4)));
typedef int      v4i __attribute__((ext_vector_type(4)));
typedef int      v8i __attribute__((ext_vector_type(8)));
typedef float    v4f __attribute__((ext_vector_type(4)));

#if defined(__AMDGCN__)
typedef __attribute__((address_space(1))) int g1int_t;   // global addrspace int
typedef __attribute__((address_space(3))) int l3int_t;   // LDS addrspace int
#endif

__device__ __forceinline__ int iclamp(int v, int lo, int hi) {
  return v < lo ? lo : (v > hi ? hi : v);
}

__global__ __launch_bounds__(NTHREADS)
void curvature_fused_kernel(const float* __restrict__ depth, float* __restrict__ out) {
  // LDS: 5440 + 4752 + 4752 = 14944 bytes / workgroup (WGP has 320 KB)
  __shared__ float sD [DROWS * DST];
  __shared__ float sGx[GROWS * GST];
  __shared__ float sGy[GROWS * GST];

  const int tid = threadIdx.x;
  const int b   = blockIdx.z;
  const int y0  = blockIdx.y * TY;
  const int x0  = blockIdx.x * TX;
  const size_t imgBase = (size_t)b * (IMG_H * IMG_W);

  // ---- Stage 1: depth halo tile (replicate semantics) into LDS ----
  const bool interior = (x0 >= 2) && (x0 + TX + 2 <= IMG_W) &&
                        (y0 >= 2) && (y0 + TY + 2 <= IMG_H);

  if (interior) {
#if defined(__AMDGCN__) && __has_builtin(__builtin_amdgcn_tensor_load_to_lds)
    // CDNA5 Tensor Data Mover: one wave issues a single 2-D tile DMA (68x20 f32),
    // tracked by TENSORcnt. EXEC is ignored by TDM, so gate at wave granularity.
    if (tid < 32) {
      uint64_t ga = (uint64_t)(uintptr_t)depth +
                    4ull * (imgBase + (size_t)(y0 - 2) * IMG_W + (size_t)(x0 - 2));
      uint32_t ldsA = (uint32_t)(uintptr_t)(&sD[0]);   // LDS aperture: low 32 bits = offset
      v4u g0;
      g0.x = 1u;                                        // count=1 (valid descriptor)
      g0.y = ldsA;                                      // lds_addr
      g0.z = (uint32_t)(ga & 0xFFFFFFFFu);              // global_addr[31:0]
      g0.w = ((uint32_t)(ga >> 32) & 0x01FFFFFFu)       // global_addr[56:32]
             | 0x80000000u;                             // type=2 ("image")
      v8i g1;
      g1[0] = 0x00020000;                               // data_size=2 (4B); wg_mask=0 (no cluster)
      g1[1] = (int)(DST << 16);                         // tensor_dim0 = 68 (lo 16 bits)
      g1[2] = (int)(DROWS << 16);                       // tensor_dim1 = 20 (lo 16 bits)
      g1[3] = (int)(DST << 16);                         // tile_dim0 = 68
      g1[4] = DROWS;                                    // tile_dim1 = 20, tile_dim2 = 0
      g1[5] = IMG_W;                                    // tensor_dim0_stride = 512 (lo 32 bits)
      g1[6] = 0;
      g1[7] = 0;
      v4i gz = {0, 0, 0, 0};
#if __has_include(<hip/amd_detail/amd_gfx1250_TDM.h>)
      v8i gz8 = {0, 0, 0, 0, 0, 0, 0, 0};
      __builtin_amdgcn_tensor_load_to_lds(g0, g1, gz, gz, gz8, 0);     // clang-23 6-arg form
#else
      __builtin_amdgcn_tensor_load_to_lds(g0, g1, gz, gz, 0);          // ROCm 7.2 5-arg form
#endif
#if __has_builtin(__builtin_amdgcn_s_wait_tensorcnt)
      __builtin_amdgcn_s_wait_tensorcnt((short)0);
#else
      asm volatile("s_wait_tensorcnt 0x0" ::: "memory");
#endif
    }
#else
    for (int idx = tid; idx < DROWS * DST; idx += NTHREADS) {
      int ry = idx / DST, rx = idx - ry * DST;
      sD[idx] = depth[imgBase + (size_t)(y0 - 2 + ry) * IMG_W + (x0 - 2 + rx)];
    }
#endif
  } else {
    // Border tiles: TDM zero-fills OOB, but we need replicate padding -> per-lane
    // clamped addresses. Async global->LDS copy (ASYNCcnt) when available.
#if defined(__AMDGCN__) && __has_builtin(__builtin_amdgcn_global_load_async_to_lds_b32)
    for (int idx = tid; idx < DROWS * DST; idx += NTHREADS) {
      int ry = idx / DST, rx = idx - ry * DST;
      int gy_ = iclamp(y0 - 2 + ry, 0, IMG_H - 1);
      int gx_ = iclamp(x0 - 2 + rx, 0, IMG_W - 1);
      const float* src = &depth[imgBase + (size_t)gy_ * IMG_W + gx_];
      __builtin_amdgcn_global_load_async_to_lds_b32(
          (g1int_t*)(uintptr_t)src, (l3int_t*)(uintptr_t)&sD[idx], 0, 0);
    }
#if __has_builtin(__builtin_amdgcn_s_wait_asynccnt)
    __builtin_amdgcn_s_wait_asynccnt(0);
#else
    asm volatile("s_wait_asynccnt 0x0" ::: "memory");
#endif
#else
    for (int idx = tid; idx < DROWS * DST; idx += NTHREADS) {
      int ry = idx / DST, rx = idx - ry * DST;
      int gy_ = iclamp(y0 - 2 + ry, 0, IMG_H - 1);
      int gx_ = iclamp(x0 - 2 + rx, 0, IMG_W - 1);
      sD[idx] = depth[imgBase + (size_t)gy_ * IMG_W + gx_];
    }
#endif
  }
  __syncthreads();

  // ---- Stage 2: first-order Sobel gx/gy over (TX+2)x(TY+2), replicate-correct halo ----
  // grad_x padded with 'edge' == grad_x evaluated at the image-clamped center, where
  // grad_x(c) itself reads clamped depth: sD[g] already holds depth(clamp(g)).
  for (int idx = tid; idx < GROWS * GST; idx += NTHREADS) {
    int ry = idx / GST, rx = idx - ry * GST;
    int cy = iclamp(y0 - 1 + ry, 0, IMG_H - 1);      // clamped global center
    int cx = iclamp(x0 - 1 + rx, 0, IMG_W - 1);
    int dy = cy - (y0 - 2);                          // 1 .. DROWS-2
    int dx = cx - (x0 - 2);                          // 1 .. DST-2
    const float* rm = &sD[(dy - 1) * DST + (dx - 1)];
    const float* rc = &sD[(dy    ) * DST + (dx - 1)];
    const float* rp = &sD[(dy + 1) * DST + (dx - 1)];
    float a00 = rm[0], a01 = rm[1], a02 = rm[2];
    float a10 = rc[0],               a12 = rc[2];
    float a20 = rp[0], a21 = rp[1], a22 = rp[2];
    sGx[idx] = (a02 - a00) + 2.0f * (a12 - a10) + (a22 - a20);           // SOBEL_X (corr.)
    sGy[idx] = (a20 + 2.0f * a21 + a22) - (a00 + 2.0f * a01 + a02);      // SOBEL_Y (corr.)
  }
  __syncthreads();

  // ---- Stage 3: second-order terms + curvature math; 4 pixels/thread, b128 NT stores ----
  const int tx = (tid & 15) << 2;     // 0..60, 16B-aligned output x
  const int ty = tid >> 4;            // 0..15
  const int rr = ty + 1;              // grad-LDS row of this output row
  const int cb = tx;                  // grad-LDS col of (center-1) for pixel 0

  float xm[6], xc[6], xp[6], ym[6], yp[6], yc[4];
#pragma unroll
  for (int i = 0; i < 6; ++i) {
    xm[i] = sGx[(rr - 1) * GST + cb + i];
    xc[i] = sGx[(rr    ) * GST + cb + i];
    xp[i] = sGx[(rr + 1) * GST + cb + i];
    ym[i] = sGy[(rr - 1) * GST + cb + i];
    yp[i] = sGy[(rr + 1) * GST + cb + i];
  }
#pragma unroll
  for (int i = 0; i < 4; ++i) yc[i] = sGy[rr * GST + cb + 1 + i];

  float res[10][4];
#pragma unroll
  for (int j = 0; j < 4; ++j) {
    float gxv = xc[j + 1];
    float gyv = yc[j];
    float gxx = (xm[j+2] - xm[j]) + 2.0f * (xc[j+2] - xc[j]) + (xp[j+2] - xp[j]);
    float gxy = (xp[j] + 2.0f * xp[j+1] + xp[j+2]) - (xm[j] + 2.0f * xm[j+1] + xm[j+2]);
    float gyy = (yp[j] + 2.0f * yp[j+1] + yp[j+2]) - (ym[j] + 2.0f * ym[j+1] + ym[j+2]);

    float inv  = rsqrtf(gxv * gxv + gyv * gyv + 1.0f);   // ||(-gx,-gy,1)|| >= 1 > eps
    float tr   = gxx + gyy;
    float det  = gxx * gyy - gxy * gxy;
    float disc = sqrtf(fmaxf(tr * tr - 4.0f * det, 1e-6f));
    float k1   = 0.5f * (tr + disc);
    float k2   = 0.5f * (tr - disc);
    float diff = k2 - k1;
    diff = (fabsf(diff) < 1e-6f) ? 1e-6f : diff;
    float si = 0.6366198f * atanf((k1 + k2) / diff);     // 2 / 3.1415926

    res[0][j] = -gxv * inv;   // normal x
    res[1][j] = -gyv * inv;   // normal y
    res[2][j] =  inv;         // normal z
    res[3][j] =  gxv;         // grad_x
    res[4][j] =  gyv;         // grad_y
    res[5][j] =  k1;
    res[6][j] =  k2;
    res[7][j] =  k1 * k2;     // gaussian curvature
    res[8][j] =  0.5f * tr;   // mean curvature
    res[9][j] =  si;          // shape index
  }

  const int yy = y0 + ty;
  const int xx = x0 + tx;
#pragma unroll
  for (int ch = 0; ch < 10; ++ch) {
    v4f v = {res[ch][0], res[ch][1], res[ch][2], res[ch][3]};
    v4f* dst = reinterpret_cast<v4f*>(
        &out[(((size_t)b * 10 + ch) * IMG_H + yy) * IMG_W + xx]);
    // Output is written once, never re-read (320 MB stream): non-temporal keeps the
    // 192 MB L2 free for the depth image (32 MB, fully resident) and its halo re-reads.
    __builtin_nontemporal_store(v, dst);
  }
}

extern "C" void kernel_launch(void* const* d_in, const int* in_sizes, int n_in,
                              void* d_out, int out_size, void* d_ws, size_t ws_size,
                              hipStream_t stream) {
  (void)in_sizes; (void)n_in; (void)out_size; (void)d_ws; (void)ws_size;
  const float* depth = (const float*)d_in[0];
  float* out = (float*)d_out;
  dim3 grid(IMG_W / TX, IMG_H / TY, NBATCH);   // 8 x 32 x 32 = 8192 workgroups
  curvature_fused_kernel<<<grid, NTHREADS, 0, stream>>>(depth, out);
}